// CMoALoRA2B_Selector_64390149701866
// MI455X (gfx1250) — compile-verified
//
#include <hip/hip_runtime.h>
#include <math.h>

// Problem constants (from reference)
#define BZ   64
#define SEQ  2048
#define DIM  2048
#define NEXP 16
#define RK   4

// SEQ-reduction split (deterministic two-stage tree, no float atomics)
#define SPLIT 8
#define CHUNK (SEQ / SPLIT)   // 256

typedef __attribute__((ext_vector_type(2))) float v2f;
typedef __attribute__((ext_vector_type(8))) float v8f;

// ---------------------------------------------------------------------------
// Stage 1: partial sums over SEQ. Each thread owns 4 consecutive dims (float4,
// 128-bit coalesced global_load_b128) and CHUNK seq positions. 128x8 blocks of
// 256 threads = 8192 wave32s -> saturates HBM. Writes SPLIT partial planes.
// ---------------------------------------------------------------------------
__global__ __launch_bounds__(256)
void reduce_seq_kernel(const float* __restrict__ x, float* __restrict__ partials) {
    const int col4 = blockIdx.x * 256 + threadIdx.x;          // 0 .. BZ*DIM/4-1
    const int d    = (col4 * 4) & (DIM - 1);
    const int b    = (col4 * 4) >> 11;                        // DIM = 2^11
    const int s0   = blockIdx.y * CHUNK;

    const float4* p = (const float4*)(x + ((size_t)b * SEQ + s0) * DIM + d);
    float4 acc = make_float4(0.f, 0.f, 0.f, 0.f);
#pragma unroll 4
    for (int s = 0; s < CHUNK; ++s) {
        float4 v = p[(size_t)s * (DIM / 4)];
        acc.x += v.x; acc.y += v.y; acc.z += v.z; acc.w += v.w;
    }
    float4* o = (float4*)(partials + (size_t)blockIdx.y * (BZ * DIM)) + col4;
    *o = acc;
}

// ---------------------------------------------------------------------------
// Stage 2: combine the SPLIT partial planes -> x_sum[b][d] (deterministic).
// ---------------------------------------------------------------------------
__global__ __launch_bounds__(256)
void reduce_partials_kernel(const float* __restrict__ partials, float* __restrict__ xsum) {
    const int col = blockIdx.x * 256 + threadIdx.x;           // 0 .. BZ*DIM-1
    float a = 0.f;
#pragma unroll
    for (int y = 0; y < SPLIT; ++y)
        a += partials[(size_t)y * (BZ * DIM) + col];
    xsum[col] = a;
}

// ---------------------------------------------------------------------------
// Stage 3: one wave32 per 16-batch tile.
//   C1 = x_sum_tile[16xK] * W_b^T[Kx16]   via V_WMMA_F32_16X16X4_F32
//   C2 = (sum_r embed[idx])[16xK] * W_c^T via V_WMMA_F32_16X16X4_F32
// Fragment layout per CDNA5 ISA (32-bit 16x4 A / 4x16 B):
//   lane 0-15  : M(=row)=lane,    K = {0,1} in vgpr {0,1}
//   lane 16-31 : M(=row)=lane-16, K = {2,3} in vgpr {0,1}
// C/D: lanes 0-15 -> M = vgpr, lanes 16-31 -> M = vgpr+8, N = lane&15.
// Epilogue (per batch row, lanes 0-15): softmax1 + softmax2, combine,
// top-4 (masked argmax, first-index tie-break), score softmax, scatter-mask
// normalize with eps clip -- exactly the reference select().
// ---------------------------------------------------------------------------
__global__ __launch_bounds__(32)
void selector_wmma_kernel(const float* __restrict__ xsum,
                          const int*   __restrict__ idxs,
                          const float* __restrict__ Wb,
                          const float* __restrict__ embed,
                          const float* __restrict__ Wc,
                          float*       __restrict__ out) {
    const int lane = threadIdx.x;       // 0..31, all active (EXEC = all 1s)
    const int m    = lane & 15;         // row within 16x16 tile (A) / col N (B)
    const int half = lane >> 4;         // K-half selector
    const int koff = half * 2;          // K offset {0,2} within each K=4 step
    const int b    = blockIdx.x * 16 + m;

    const float* xrow  = xsum  + (size_t)b * DIM + koff;
    const float* wbrow = Wb    + (size_t)m * DIM + koff;   // B[K][N=m] = W_b[m][K]
    const float* wcrow = Wc    + (size_t)m * DIM + koff;
    const int4   ei    = *(const int4*)(idxs + b * RK);
    const float* e0 = embed + (size_t)ei.x * DIM + koff;
    const float* e1 = embed + (size_t)ei.y * DIM + koff;
    const float* e2 = embed + (size_t)ei.z * DIM + koff;
    const float* e3 = embed + (size_t)ei.w * DIM + koff;

    v8f c1 = {};
    v8f c2 = {};
#pragma unroll 4
    for (int k0 = 0; k0 < DIM; k0 += 4) {
        v2f a  = *(const v2f*)(xrow  + k0);
        v2f bb = *(const v2f*)(wbrow + k0);
        v2f bc = *(const v2f*)(wcrow + k0);
        v2f h;
        h[0] = (e0[k0]     + e1[k0])     + (e2[k0]     + e3[k0]);
        h[1] = (e0[k0 + 1] + e1[k0 + 1]) + (e2[k0 + 1] + e3[k0 + 1]);
        c1 = __builtin_amdgcn_wmma_f32_16x16x4_f32(false, a, false, bb, (short)0, c1, false, false);
        c2 = __builtin_amdgcn_wmma_f32_16x16x4_f32(false, h, false, bc, (short)0, c2, false, false);
    }

    __shared__ float l1s[16][16];
    __shared__ float l2s[16][16];
    {
        const int mb = half * 8;
#pragma unroll
        for (int v = 0; v < 8; ++v) {
            l1s[mb + v][m] = c1[v] * (1.0f / (float)SEQ);  // mean over SEQ
            l2s[mb + v][m] = c2[v] * (1.0f / (float)RK);   // mean over R
        }
    }
    __syncthreads();

    if (lane < 16) {
        const int row = lane;
        const int bout = blockIdx.x * 16 + row;

        float p1[16], p2[16], lg[16];
        float mx1 = -INFINITY, mx2 = -INFINITY;
#pragma unroll
        for (int e = 0; e < NEXP; ++e) {
            mx1 = fmaxf(mx1, l1s[row][e]);
            mx2 = fmaxf(mx2, l2s[row][e]);
        }
        float s1 = 0.f, s2 = 0.f;
#pragma unroll
        for (int e = 0; e < NEXP; ++e) {
            p1[e] = __expf(l1s[row][e] - mx1); s1 += p1[e];
            p2[e] = __expf(l2s[row][e] - mx2); s2 += p2[e];
        }
        const float r1 = 1.f / s1, r2 = 1.f / s2;
#pragma unroll
        for (int e = 0; e < NEXP; ++e)
            lg[e] = p1[e] * r1 + p2[e] * r2;               // IFS_WEIGHT = 1.0

        // top-4: iterated masked argmax (descending, first-index tie-break)
        unsigned taken = 0u;
        int pick[RK];
#pragma unroll
        for (int r = 0; r < RK; ++r) {
            float best = -INFINITY; int bi = 0;
#pragma unroll
            for (int e = 0; e < NEXP; ++e) {
                if (!((taken >> e) & 1u) && lg[e] > best) { best = lg[e]; bi = e; }
            }
            taken |= (1u << bi);
            pick[r] = bi;
        }

        // score = softmax(lg); masked + eps-clipped normalize
        float mg = -INFINITY;
#pragma unroll
        for (int e = 0; e < NEXP; ++e) mg = fmaxf(mg, lg[e]);
        float sc[16]; float ssum = 0.f;
#pragma unroll
        for (int e = 0; e < NEXP; ++e) { sc[e] = __expf(lg[e] - mg); ssum += sc[e]; }
        float psum = 0.f;
#pragma unroll
        for (int e = 0; e < NEXP; ++e) if ((taken >> e) & 1u) psum += sc[e];
        // ref: denom = max(eps, masked_sum/ssum); norm = (sc/ssum)/denom
        const float denom = fmaxf(psum, 1.1920929e-07f * ssum);
        const float rden = 1.f / denom;
#pragma unroll
        for (int e = 0; e < NEXP; ++e)
            out[bout * NEXP + e] = ((taken >> e) & 1u) ? sc[e] * rden : 0.f;
#pragma unroll
        for (int r = 0; r < RK; ++r)
            out[BZ * NEXP + bout * RK + r] = (float)pick[r];
    }
}

// ---------------------------------------------------------------------------
extern "C" void kernel_launch(void* const* d_in, const int* in_sizes, int n_in,
                              void* d_out, int out_size, void* d_ws, size_t ws_size,
                              hipStream_t stream) {
    const float* x     = (const float*)d_in[0];  // [BZ,SEQ,DIM] f32
    const int*   idx   = (const int*)  d_in[1];  // [BZ,R] i32
    const float* Wb    = (const float*)d_in[2];  // [E,DIM]
    const float* embed = (const float*)d_in[3];  // [E,DIM]
    const float* Wc    = (const float*)d_in[4];  // [E,DIM]
    float* out = (float*)d_out;                  // [BZ*E] scores + [BZ*R] indices

    float* partials = (float*)d_ws;                          // SPLIT * BZ*DIM f32 (4 MB)
    float* xsum     = partials + (size_t)SPLIT * BZ * DIM;   // BZ*DIM f32 (512 KB)

    dim3 g1(BZ * DIM / 4 / 256, SPLIT);                      // 128 x 8 blocks
    reduce_seq_kernel<<<g1, 256, 0, stream>>>(x, partials);
    reduce_partials_kernel<<<BZ * DIM / 256, 256, 0, stream>>>(partials, xsum);
    selector_wmma_kernel<<<BZ / 16, 32, 0, stream>>>(xsum, idx, Wb, embed, Wc, out);
}